// DiceLoss_54425825575360
// MI455X (gfx1250) — compile-verified
//
#include <hip/hip_runtime.h>

// Dice loss, fused single-pass streaming reduction for MI455X (gfx1250).
// Bandwidth-bound: ~176 MB read once -> ~7.6 us floor @ 23.3 TB/s.
// v2: vectorized global traffic -- float4 (global_load_b128) per class,
// 4 pixels per thread, target via longlong2 b128 loads. Softmax entirely in
// registers; per-thread partials reduced across lanes with
// V_WMMA_F32_16X16X4_F32 (ones-matrix reduction trick).

#define NUM_CLASS 19
#define NBATCH    8
#define PIX       (512 * 512)
#define SMOOTHF   1.0f
#define BLOCKS_PER_N 64
#define TPB          256

typedef __attribute__((ext_vector_type(2))) float v2f;
typedef __attribute__((ext_vector_type(8))) float v8f;

__device__ __forceinline__ float f4_get(const float4& v, int j) {
    // j is a compile-time constant after unrolling -> folds to a register read
    switch (j) {
        case 0:  return v.x;
        case 1:  return v.y;
        case 2:  return v.z;
        default: return v.w;
    }
}

// Sum a float across all 32 lanes of a wave using the matrix pipe.
// D = ones(16x4) x B(4x16), with both B VGPRs holding x: column n of B only
// touches lanes n and n+16, so d[0] lane n = 2*(x[n&15] + x[(n&15)+16]).
// Four xor-shuffles finish the 16 column sums; scale by 0.5.
// Requires EXEC all-ones (call only from fully converged code).
__device__ __forceinline__ float wave_sum_wmma(float x) {
    v2f a; a.x = 1.0f; a.y = 1.0f;   // A = ones(16x4)
    v2f b; b.x = x;    b.y = x;      // B = data, duplicated across both VGPRs
    v8f c = {};
    v8f d = __builtin_amdgcn_wmma_f32_16x16x4_f32(
        /*neg_a=*/false, a, /*neg_b=*/false, b,
        /*c_mod=*/(short)0, c, /*reuse_a=*/false, /*reuse_b=*/false);
    float y = d[0];
    y += __shfl_xor(y, 1, 32);
    y += __shfl_xor(y, 2, 32);
    y += __shfl_xor(y, 4, 32);
    y += __shfl_xor(y, 8, 32);
    return 0.5f * y;
}

__global__ void dice_init(float* __restrict__ ws) {
    int i = threadIdx.x;
    if (i < 3 * NBATCH * NUM_CLASS) ws[i] = 0.0f;
}

__global__ __launch_bounds__(TPB) void dice_partial(
        const float* __restrict__ predict,
        const long long* __restrict__ target,
        float* __restrict__ ws) {
    const int n   = blockIdx.y;
    const int tid = threadIdx.x;

    float* interW = ws;
    float* psumW  = ws + NBATCH * NUM_CLASS;
    float* tsumW  = ws + 2 * NBATCH * NUM_CLASS;

    float psAcc[NUM_CLASS], inAcc[NUM_CLASS], tAcc[NUM_CLASS];
#pragma unroll
    for (int c = 0; c < NUM_CLASS; ++c) { psAcc[c] = 0.f; inAcc[c] = 0.f; tAcc[c] = 0.f; }

    const float* pbase = predict + (size_t)n * NUM_CLASS * PIX;
    const longlong2* tbase2 =
        reinterpret_cast<const longlong2*>(target + (size_t)n * PIX);

    const int npix4  = PIX / 4;               // 65536 float4 groups
    const int stride = BLOCKS_PER_N * TPB;    // 16384; npix4/stride == 4 (uniform)

    for (int p4 = blockIdx.x * TPB + tid; p4 < npix4; p4 += stride) {
        // Speculative prefetch of next iteration's plane-0 line (gfx1250
        // global_prefetch_b8; translation failures silently dropped).
        __builtin_prefetch(pbase + (size_t)(p4 + stride) * 4, 0, 1);

        // 4 int64 targets for these 4 pixels: two b128 loads
        const longlong2 t01 = tbase2[2 * p4];
        const longlong2 t23 = tbase2[2 * p4 + 1];
        int ti[4];
        ti[0] = (int)t01.x; ti[1] = (int)t01.y;
        ti[2] = (int)t23.x; ti[3] = (int)t23.y;

        // 19 class planes, b128 each (coalesced: 512B per wave per plane)
        float4 x4[NUM_CLASS];
#pragma unroll
        for (int c = 0; c < NUM_CLASS; ++c)
            x4[c] = reinterpret_cast<const float4*>(pbase + (size_t)c * PIX)[p4];

#pragma unroll
        for (int j = 0; j < 4; ++j) {
            float x[NUM_CLASS];
#pragma unroll
            for (int c = 0; c < NUM_CLASS; ++c) x[c] = f4_get(x4[c], j);

            float m = x[0];
#pragma unroll
            for (int c = 1; c < NUM_CLASS; ++c) m = fmaxf(m, x[c]);

            float denom = 0.0f;
#pragma unroll
            for (int c = 0; c < NUM_CLASS; ++c) { x[c] = __expf(x[c] - m); denom += x[c]; }

            const float inv = __builtin_amdgcn_rcpf(denom);
            const int   t   = ti[j];
#pragma unroll
            for (int c = 0; c < NUM_CLASS; ++c) {
                const float p  = x[c] * inv;
                const bool hit = (t == c);
                psAcc[c] += p;
                inAcc[c] += hit ? p : 0.0f;
                tAcc[c]  += hit ? 1.0f : 0.0f;
            }
        }
    }

    __shared__ float sacc[3 * NUM_CLASS];
    if (tid < 3 * NUM_CLASS) sacc[tid] = 0.0f;
    __syncthreads();

    const int lane = tid & 31;
#pragma unroll
    for (int c = 0; c < NUM_CLASS; ++c) {
        const float vI = wave_sum_wmma(inAcc[c]);   // v_wmma_f32_16x16x4_f32
        const float vP = wave_sum_wmma(psAcc[c]);
        const float vT = wave_sum_wmma(tAcc[c]);
        if (lane == 0) {
            atomicAdd(&sacc[c],                 vI);   // ds_add_f32
            atomicAdd(&sacc[NUM_CLASS + c],     vP);
            atomicAdd(&sacc[2 * NUM_CLASS + c], vT);
        }
    }
    __syncthreads();

    if (tid < NUM_CLASS) {
        atomicAdd(&interW[n * NUM_CLASS + tid], sacc[tid]);                  // global_atomic_add_f32
        atomicAdd(&psumW [n * NUM_CLASS + tid], sacc[NUM_CLASS + tid]);
        atomicAdd(&tsumW [n * NUM_CLASS + tid], sacc[2 * NUM_CLASS + tid]);
    }
}

__global__ void dice_finalize(const float* __restrict__ ws, float* __restrict__ out) {
    const float* interW = ws;
    const float* psumW  = ws + NBATCH * NUM_CLASS;
    const float* tsumW  = ws + 2 * NBATCH * NUM_CLASS;

    const int c = threadIdx.x;   // one wave: lanes 0..18 carry classes
    float acc = 0.0f;
    if (c < NUM_CLASS) {
#pragma unroll
        for (int n = 0; n < NBATCH; ++n) {
            const float I = interW[n * NUM_CLASS + c];
            const float P = psumW [n * NUM_CLASS + c];
            const float T = tsumW [n * NUM_CLASS + c];
            acc += 1.0f - (2.0f * I + SMOOTHF) / (P + T + SMOOTHF);
        }
        acc *= (1.0f / NBATCH);   // mean over batch
    }
    acc += __shfl_xor(acc, 1, 32);
    acc += __shfl_xor(acc, 2, 32);
    acc += __shfl_xor(acc, 4, 32);
    acc += __shfl_xor(acc, 8, 32);
    acc += __shfl_xor(acc, 16, 32);
    if (threadIdx.x == 0) out[0] = acc / (float)NUM_CLASS;
}

extern "C" void kernel_launch(void* const* d_in, const int* in_sizes, int n_in,
                              void* d_out, int out_size, void* d_ws, size_t ws_size,
                              hipStream_t stream) {
    const float*     predict = (const float*)d_in[0];      // [8,19,512,512] fp32
    const long long* target  = (const long long*)d_in[1];  // [8,512,512] int64
    float* ws  = (float*)d_ws;                             // 3 * 8 * 19 floats
    float* out = (float*)d_out;                            // 1 float

    dice_init<<<dim3(1), dim3(512), 0, stream>>>(ws);

    dim3 grid(BLOCKS_PER_N, NBATCH);
    dice_partial<<<grid, dim3(TPB), 0, stream>>>(predict, target, ws);

    dice_finalize<<<dim3(1), dim3(32), 0, stream>>>(ws, out);
}